// RGCNEncoder_16870631539383
// MI455X (gfx1250) — compile-verified
//
#include <hip/hip_runtime.h>

// ---------------------------------------------------------------------------
// RGCN encoder (2x FastRGCNConv, basis decomposition, per-(dst,rel) mean aggr)
// for MI455X / gfx1250.  FP32 WMMA (v_wmma_f32_16x16x4_f32) path.
// LDS W tile stored TRANSPOSED so B fragments are contiguous ds_load_b64.
// ---------------------------------------------------------------------------

#define N_NODES   100000
#define HIDDEN    128
#define N_REL     64
#define N_BASES   30
#define N_EDGES   600000
#define XPAD      132              // padded LDS row stride (dwords); 132 % 64 == 4

typedef float v2f __attribute__((ext_vector_type(2)));
typedef float v8f __attribute__((ext_vector_type(8)));

// ---------------------------------------------------------------------------
// Edge bucketing: histogram, scan, scatter, per-edge norm
// ---------------------------------------------------------------------------
__global__ void k_hist(const int* __restrict__ ei, const int* __restrict__ et,
                       int* __restrict__ counts, int* __restrict__ cnt) {
  int e = blockIdx.x * blockDim.x + threadIdx.x;
  if (e >= N_EDGES) return;
  int r = et[e];
  int d = ei[N_EDGES + e];              // edge_index[1] = dst
  atomicAdd(&counts[r], 1);
  atomicAdd(&cnt[d * N_REL + r], 1);
}

__global__ void k_scan(const int* __restrict__ counts, int* __restrict__ offsets) {
  int acc = 0;
  for (int r = 0; r < N_REL; ++r) { offsets[r] = acc; acc += counts[r]; }
  offsets[N_REL] = acc;
}

__global__ void k_scatter(const int* __restrict__ et, const int* __restrict__ offsets,
                          int* __restrict__ cursor, int* __restrict__ perm) {
  int e = blockIdx.x * blockDim.x + threadIdx.x;
  if (e >= N_EDGES) return;
  int r = et[e];
  int p = atomicAdd(&cursor[r], 1);
  perm[offsets[r] + p] = e;
}

__global__ void k_norm(const int* __restrict__ ei, const int* __restrict__ et,
                       const int* __restrict__ cnt, float* __restrict__ normE) {
  int e = blockIdx.x * blockDim.x + threadIdx.x;
  if (e >= N_EDGES) return;
  int r = et[e];
  int d = ei[N_EDGES + e];
  int c = cnt[d * N_REL + r];
  normE[e] = 1.0f / (float)(c < 1 ? 1 : c);
}

// ---------------------------------------------------------------------------
// W[r] = sum_b comp[r,b] * basis[b]   (tiny: 63 MFLOP, plain VALU)
// ---------------------------------------------------------------------------
__global__ void k_weights(const float* __restrict__ comp, const float* __restrict__ basis,
                          float* __restrict__ W) {
  int idx = blockIdx.x * blockDim.x + threadIdx.x;
  if (idx >= N_REL * HIDDEN * HIDDEN) return;
  int r  = idx >> 14;                   // / 16384
  int io = idx & 16383;
  float acc = 0.f;
#pragma unroll
  for (int b = 0; b < N_BASES; ++b)
    acc = fmaf(comp[r * N_BASES + b], basis[b * HIDDEN * HIDDEN + io], acc);
  W[idx] = acc;
}

// Stage a 128x128 f32 weight matrix into LDS, transposed (Wl[n*XPAD + k] = W[k][n]).
// Global reads stay float4-coalesced; transpose cost is amortized (once per block).
__device__ __forceinline__ void stage_w_transposed(const float* __restrict__ Wg,
                                                   float* __restrict__ Wl, int tid) {
  for (int i = tid; i < HIDDEN * (HIDDEN / 4); i += 256) {
    float4 v = reinterpret_cast<const float4*>(Wg)[i];
    int k  = i >> 5;                    // input-dim row in global
    int c4 = (i & 31) << 2;             // output-dim col base
    Wl[(c4 + 0) * XPAD + k] = v.x;
    Wl[(c4 + 1) * XPAD + k] = v.y;
    Wl[(c4 + 2) * XPAD + k] = v.z;
    Wl[(c4 + 3) * XPAD + k] = v.w;
  }
}

// ---------------------------------------------------------------------------
// Per-relation message GEMM with WMMA.
//   grid = (N_REL, CHUNKS), block = 256 (8 waves).  Wave handles one 16-edge
//   M-tile x full 128 outputs: 32 k-steps x 8 n-tiles of v_wmma_f32_16x16x4_f32.
//   B (= W_r^T) staged once per block in LDS; A rows gathered per tile.
//   Output: out[dst] += norm_e * (x[src] @ W_r)   via global_atomic_add_f32.
// ---------------------------------------------------------------------------
__global__ __launch_bounds__(256) void k_msg(
    const float* __restrict__ x, const float* __restrict__ W,
    const int* __restrict__ ei, const int* __restrict__ perm,
    const int* __restrict__ offsets, const float* __restrict__ normE,
    float* __restrict__ out) {
  extern __shared__ float smem[];
  float* Wl = smem;                                 // 128 x XPAD (transposed W_r)
  float* Xl = Wl + HIDDEN * XPAD;                   // 8 waves * 16 x XPAD
  float* nl = Xl + 8 * 16 * XPAD;                   // 8 * 16 norms
  int*   dl = (int*)(nl + 8 * 16);                  // 8 * 16 dst rows

  const int r      = blockIdx.x;
  const int segBeg = offsets[r];
  const int segEnd = offsets[r + 1];
  if (segEnd == segBeg) return;
  const int nTiles = (segEnd - segBeg + 15) >> 4;

  const int tid  = threadIdx.x;
  const int wave = tid >> 5;
  const int lane = tid & 31;
  const int m    = lane & 15;                       // A row / D col index
  const int kb   = (lane >> 4) << 1;                // 0 or 2: K sub-pair per lane half

  stage_w_transposed(W + (size_t)r * HIDDEN * HIDDEN, Wl, tid);
  __syncthreads();

  float* Xw = Xl + wave * 16 * XPAD;
  float* nw = nl + wave * 16;
  int*   dw = dl + wave * 16;

  for (int t = blockIdx.y * 8 + wave; t < nTiles; t += gridDim.y * 8) {
    // Per-edge metadata for this 16-edge tile (tail edges padded with norm=0).
    int pos = segBeg + t * 16 + m;
    int sv = 0, dv = 0; float nv = 0.f;
    if (pos < segEnd) {
      int e = perm[pos];
      sv = ei[e];                                   // src
      dv = ei[N_EDGES + e];                         // dst
      nv = normE[e];
    }
    if (lane < 16) { nw[m] = nv; dw[m] = dv; }

    // Gather the 16 source rows into this wave's LDS region (1 row / iter).
    for (int rr = 0; rr < 16; ++rr) {
      int srow = __shfl(sv, rr);
      float4 v = reinterpret_cast<const float4*>(x + (size_t)srow * HIDDEN)[lane];
      float* dp = &Xw[rr * XPAD + (lane << 2)];
      dp[0] = v.x; dp[1] = v.y; dp[2] = v.z; dp[3] = v.w;
    }
    asm volatile("s_wait_dscnt 0" ::: "memory");    // cross-lane LDS visibility

    v8f acc[8];
#pragma unroll
    for (int nt = 0; nt < 8; ++nt) acc[nt] = v8f{};

    for (int ks = 0; ks < 32; ++ks) {               // K = 128 in steps of 4
      int k = (ks << 2) + kb;                       // even -> 8B aligned ds_load_b64
      v2f a = *reinterpret_cast<const v2f*>(&Xw[m * XPAD + k]);
#pragma unroll
      for (int nt = 0; nt < 8; ++nt) {
        v2f b = *reinterpret_cast<const v2f*>(&Wl[((nt << 4) + m) * XPAD + k]);
        acc[nt] = __builtin_amdgcn_wmma_f32_16x16x4_f32(
            false, a, false, b, (short)0, acc[nt], false, false);
      }
    }

    // D layout: lane holds col n = m, rows j + 8*(lane>>4).  Scale + scatter.
#pragma unroll
    for (int j = 0; j < 8; ++j) {
      int mrow = j + ((lane >> 4) << 3);
      float nv2 = nw[mrow];
      float* op = out + (size_t)dw[mrow] * HIDDEN + m;
#pragma unroll
      for (int nt = 0; nt < 8; ++nt)
        atomicAdd(op + (nt << 4), acc[nt][j] * nv2);
    }
  }
}

// ---------------------------------------------------------------------------
// Root transform: inout[n] = (relu?)( inout[n] + x[n] @ rootW + bias )
// Same WMMA tile structure; non-atomic in-place epilogue.
// ---------------------------------------------------------------------------
__global__ __launch_bounds__(256) void k_root(
    const float* __restrict__ x, const float* __restrict__ rootW,
    const float* __restrict__ bias, float* __restrict__ inout, int doRelu) {
  extern __shared__ float smem[];
  float* Wl = smem;                                 // 128 x XPAD (transposed root)
  float* Xl = Wl + HIDDEN * XPAD;                   // 8 * 16 x XPAD
  float* bl = Xl + 8 * 16 * XPAD;                   // 128

  const int tid  = threadIdx.x;
  const int wave = tid >> 5;
  const int lane = tid & 31;
  const int m    = lane & 15;
  const int kb   = (lane >> 4) << 1;

  stage_w_transposed(rootW, Wl, tid);
  if (tid < HIDDEN) bl[tid] = bias[tid];
  __syncthreads();

  float* Xw = Xl + wave * 16 * XPAD;
  const int nTiles = (N_NODES + 15) >> 4;

  for (int t = blockIdx.x * 8 + wave; t < nTiles; t += gridDim.x * 8) {
    int baseRow = t << 4;
    for (int rr = 0; rr < 16; ++rr) {
      int row = baseRow + rr; if (row >= N_NODES) row = 0;
      float4 v = reinterpret_cast<const float4*>(x + (size_t)row * HIDDEN)[lane];
      float* dp = &Xw[rr * XPAD + (lane << 2)];
      dp[0] = v.x; dp[1] = v.y; dp[2] = v.z; dp[3] = v.w;
    }
    asm volatile("s_wait_dscnt 0" ::: "memory");

    v8f acc[8];
#pragma unroll
    for (int nt = 0; nt < 8; ++nt) acc[nt] = v8f{};

    for (int ks = 0; ks < 32; ++ks) {
      int k = (ks << 2) + kb;
      v2f a = *reinterpret_cast<const v2f*>(&Xw[m * XPAD + k]);
#pragma unroll
      for (int nt = 0; nt < 8; ++nt) {
        v2f b = *reinterpret_cast<const v2f*>(&Wl[((nt << 4) + m) * XPAD + k]);
        acc[nt] = __builtin_amdgcn_wmma_f32_16x16x4_f32(
            false, a, false, b, (short)0, acc[nt], false, false);
      }
    }

#pragma unroll
    for (int j = 0; j < 8; ++j) {
      int row = baseRow + j + ((lane >> 4) << 3);
      if (row >= N_NODES) continue;
      float* op = inout + (size_t)row * HIDDEN + m;
#pragma unroll
      for (int nt = 0; nt < 8; ++nt) {
        float v = op[nt << 4] + acc[nt][j] + bl[(nt << 4) + m];
        if (doRelu) v = fmaxf(v, 0.f);
        op[nt << 4] = v;
      }
    }
  }
}

// ---------------------------------------------------------------------------
// Host orchestration (graph-capture safe: only kernels + hipMemsetAsync).
// ---------------------------------------------------------------------------
extern "C" void kernel_launch(void* const* d_in, const int* in_sizes, int n_in,
                              void* d_out, int out_size, void* d_ws, size_t ws_size,
                              hipStream_t stream) {
  const int*   ei     = (const int*)d_in[0];     // [2, E]
  const int*   et     = (const int*)d_in[1];     // [E]
  const float* x0     = (const float*)d_in[2];   // [N, 128]
  const float* comp1  = (const float*)d_in[3];
  const float* basis1 = (const float*)d_in[4];
  const float* root1  = (const float*)d_in[5];
  const float* bias1  = (const float*)d_in[6];
  const float* comp2  = (const float*)d_in[7];
  const float* basis2 = (const float*)d_in[8];
  const float* root2  = (const float*)d_in[9];
  const float* bias2  = (const float*)d_in[10];
  float* outp = (float*)d_out;

  // Workspace partition (~82 MiB total, all 16B-aligned).
  char* ws = (char*)d_ws;
  float* Wbuf    = (float*)ws;  ws += (size_t)N_REL * HIDDEN * HIDDEN * 4;
  int*   cnt     = (int*)ws;    ws += (size_t)N_NODES * N_REL * 4;
  int*   counts  = (int*)ws;    ws += 64 * 4;
  int*   offsets = (int*)ws;    ws += 80 * 4;       // 65 used, padded
  int*   cursor  = (int*)ws;    ws += 64 * 4;
  int*   perm    = (int*)ws;    ws += (size_t)N_EDGES * 4;
  float* normE   = (float*)ws;  ws += (size_t)N_EDGES * 4;
  float* x1      = (float*)ws;  ws += (size_t)N_NODES * HIDDEN * 4;

  const int EB = (N_EDGES + 255) / 256;
  const int WB = (N_REL * HIDDEN * HIDDEN + 255) / 256;
  const size_t msgLds  = ((size_t)HIDDEN * XPAD + 8 * 16 * XPAD + 8 * 16) * 4 + 8 * 16 * 4;
  const size_t rootLds = ((size_t)HIDDEN * XPAD + 8 * 16 * XPAD + HIDDEN) * 4;

  // ---- edge bucketing + norms (shared by both layers) ----
  hipMemsetAsync(cnt,    0, (size_t)N_NODES * N_REL * 4, stream);
  hipMemsetAsync(counts, 0, 64 * 4, stream);
  hipMemsetAsync(cursor, 0, 64 * 4, stream);
  k_hist   <<<EB, 256, 0, stream>>>(ei, et, counts, cnt);
  k_scan   <<<1, 1, 0, stream>>>(counts, offsets);
  k_scatter<<<EB, 256, 0, stream>>>(et, offsets, cursor, perm);
  k_norm   <<<EB, 256, 0, stream>>>(ei, et, cnt, normE);

  // ---- layer 1: x1 = relu(msg_agg(x0) + x0 @ root1 + bias1) ----
  k_weights<<<WB, 256, 0, stream>>>(comp1, basis1, Wbuf);
  hipMemsetAsync(x1, 0, (size_t)N_NODES * HIDDEN * 4, stream);
  k_msg <<<dim3(N_REL, 16), 256, msgLds, stream>>>(x0, Wbuf, ei, perm, offsets, normE, x1);
  k_root<<<782, 256, rootLds, stream>>>(x0, root1, bias1, x1, 1);

  // ---- layer 2: out = msg_agg(x1) + x1 @ root2 + bias2 ----
  k_weights<<<WB, 256, 0, stream>>>(comp2, basis2, Wbuf);
  hipMemsetAsync(outp, 0, (size_t)N_NODES * HIDDEN * 4, stream);
  k_msg <<<dim3(N_REL, 16), 256, msgLds, stream>>>(x1, Wbuf, ei, perm, offsets, normE, outp);
  k_root<<<782, 256, rootLds, stream>>>(x1, root2, bias2, outp, 0);
}